// GCLMemory_36790689858236
// MI455X (gfx1250) — compile-verified
//
#include <hip/hip_runtime.h>

typedef __attribute__((ext_vector_type(2))) float v2f;
typedef __attribute__((ext_vector_type(8))) float v8f;

#define B_ 32
#define N_ 8192
#define M_ 128
#define K_ 128

// ---------------------------------------------------------------------------
// Kernel 1: s[b,n] = beta_b * cosine(keys[b,n,:], k[b,:])
// Per wave: 16 rows. Dot products via V_WMMA_F32_16X16X4_F32 accumulating over
// 32 K-chunks of 4. A = broadcast k chunk (from LDS), B = 16 key rows.
//   A 16x4 layout: lane L holds A[M=L%16][K = (L<16 ? 0,1 : 2,3)]  -> k[4i+kh], k[4i+kh+1]
//   B 4x16 layout: lane L holds B[K = (L<16 ? 0,1 : 2,3)][col=L%16] -> keys[row L%16][4i+kh..]
//   D: all rows identical (A rows identical); c[0] lane j = dot(keys row j, k).
// Row sq-norms accumulate from the same B loads; halves combined via shfl_xor 16.
// ---------------------------------------------------------------------------
__global__ __launch_bounds__(256) void k1_scores(
    const float* __restrict__ keys, const float* __restrict__ kq,
    const float* __restrict__ beta, float* __restrict__ s_buf)
{
    __shared__ float k_l[K_];
    __shared__ float red[256];
    const int t = threadIdx.x;
    const int b = blockIdx.y;

    const float kv = kq[b * K_ + (t & (K_ - 1))];
    if (t < K_) k_l[t] = kv;
    red[t] = (t < K_) ? kv * kv : 0.0f;
    __syncthreads();
    for (int off = 128; off > 0; off >>= 1) {
        if (t < off) red[t] += red[t + off];
        __syncthreads();
    }
    const float knorm = sqrtf(red[0]);

    const int wv = t >> 5;
    const int lane = t & 31;
    const int m = lane & 15;
    const int khalf = (lane < 16) ? 0 : 2;
    const int rowbase = blockIdx.x * 128 + wv * 16;
    const float* krow = keys + ((size_t)b * N_ + rowbase + m) * K_ + khalf;

    v8f c = {0.f, 0.f, 0.f, 0.f, 0.f, 0.f, 0.f, 0.f};
    float nsq = 0.0f;
#pragma unroll 4
    for (int i = 0; i < 32; ++i) {
        float2 bb = *(const float2*)(krow + i * 4);          // keys chunk (B operand)
        float2 aa = *(const float2*)(&k_l[i * 4 + khalf]);   // k chunk broadcast (A operand)
        nsq = fmaf(bb.x, bb.x, nsq);
        nsq = fmaf(bb.y, bb.y, nsq);
        v2f av; av.x = aa.x; av.y = aa.y;
        v2f bv; bv.x = bb.x; bv.y = bb.y;
        c = __builtin_amdgcn_wmma_f32_16x16x4_f32(
                /*neg_a=*/false, av, /*neg_b=*/false, bv,
                /*c_mod=*/(short)0, c, /*reuse_a=*/false, /*reuse_b=*/false);
    }
    const float dot = c[0];                       // dot of row (lane & 15)
    const float nfull = nsq + __shfl_xor(nsq, 16); // combine K%4 in {0,1} and {2,3} halves
    if (lane < 16) {
        const float norm = sqrtf(nfull);
        const float cosv = dot / (fmaxf(norm, 1e-8f) * fmaxf(knorm, 1e-8f));
        s_buf[(size_t)b * N_ + rowbase + lane] = beta[b] * cosv;
    }
}

// ---------------------------------------------------------------------------
// Kernel 2: per-batch stats. s row (32KB) resident in LDS; 4 sweeps:
//   max/argmax (first-occurrence tie-break, matching lax.top_k),
//   sumexp, masked sum -> denom, sharpen sum. Also gathers r.
// ---------------------------------------------------------------------------
__global__ __launch_bounds__(256) void k2_stats(
    const float* __restrict__ s_buf, const float* __restrict__ content,
    const float* __restrict__ gamma, float* __restrict__ params,
    float* __restrict__ r_out)
{
    __shared__ float sl[N_];
    __shared__ float rv[256];
    __shared__ int   ri[256];
    const int b = blockIdx.x;
    const int t = threadIdx.x;

    for (int i = t; i < N_; i += 256) sl[i] = s_buf[(size_t)b * N_ + i];
    __syncthreads();

    // --- max / argmax ---
    float mv = -3.402823466e38f; int mi = N_;
    for (int i = t; i < N_; i += 256) { float v = sl[i]; if (v > mv) { mv = v; mi = i; } }
    rv[t] = mv; ri[t] = mi;
    __syncthreads();
    for (int off = 128; off > 0; off >>= 1) {
        if (t < off) {
            float v = rv[t + off]; int j = ri[t + off];
            if (v > rv[t] || (v == rv[t] && j < ri[t])) { rv[t] = v; ri[t] = j; }
        }
        __syncthreads();
    }
    const float maxs = rv[0];
    const int   idx  = ri[0];
    __syncthreads();

    // --- sum of exp(s - max) ---
    float se = 0.f;
    for (int i = t; i < N_; i += 256) se += expf(sl[i] - maxs);
    rv[t] = se;
    __syncthreads();
    for (int off = 128; off > 0; off >>= 1) { if (t < off) rv[t] += rv[t + off]; __syncthreads(); }
    const float sumexp = rv[0];
    __syncthreads();
    const float inv_se = 1.0f / sumexp;

    // --- masked sum: denom = sum(wc * mask) ---
    float ms = 0.f;
    for (int i = t; i < N_; i += 256) {
        float wc = expf(sl[i] - maxs) * inv_se;
        ms += wc * ((i == idx) ? 1.0f : 1e-16f);
    }
    rv[t] = ms;
    __syncthreads();
    for (int off = 128; off > 0; off >>= 1) { if (t < off) rv[t] += rv[t + off]; __syncthreads(); }
    const float denom = rv[0];
    __syncthreads();
    const float invden = 1.0f / denom;

    // --- sharpen sum: wsum = sum((wc*mask/denom + 1e-10)^gamma) ---
    const float g = gamma[b];
    float wsu = 0.f;
    for (int i = t; i < N_; i += 256) {
        float wc = expf(sl[i] - maxs) * inv_se;
        float x  = wc * ((i == idx) ? 1.0f : 1e-16f) * invden + 1e-10f;
        wsu += exp2f(g * log2f(x));
    }
    rv[t] = wsu;
    __syncthreads();
    for (int off = 128; off > 0; off >>= 1) { if (t < off) rv[t] += rv[t + off]; __syncthreads(); }

    if (t == 0) {
        params[b * 8 + 0] = maxs;
        params[b * 8 + 1] = inv_se;
        params[b * 8 + 2] = denom;
        params[b * 8 + 3] = rv[0];          // wsum
        ((int*)params)[b * 8 + 4] = idx;
    }
    // --- r = content[b, idx, :] ---
    if (t < M_) r_out[b * M_ + t] = content[((size_t)b * N_ + idx) * M_ + t];
}

// ---------------------------------------------------------------------------
// Kernel 3: streaming pass (carries ~540MB). One wave per row: lane holds one
// float4 of the 128-float row; fully coalesced 512B per array per wave.
// w recomputed per wave from s + per-batch params.
// ---------------------------------------------------------------------------
__global__ __launch_bounds__(256) void k3_update(
    const float* __restrict__ content, const float* __restrict__ keys,
    const float* __restrict__ a, const float* __restrict__ a_k,
    const float* __restrict__ gamma, const float* __restrict__ s_buf,
    const float* __restrict__ params,
    float* __restrict__ w_out, float* __restrict__ nc_out, float* __restrict__ nk_out)
{
    const int b    = blockIdx.y;
    const int row  = blockIdx.x * 8 + (threadIdx.x >> 5);
    const int lane = threadIdx.x & 31;

    const float maxs   = params[b * 8 + 0];
    const float inv_se = params[b * 8 + 1];
    const float invden = 1.0f / params[b * 8 + 2];
    const float wnorm  = 1.0f / (params[b * 8 + 3] + 1e-10f);
    const int   idx    = ((const int*)params)[b * 8 + 4];
    const float g      = gamma[b];

    const float s  = s_buf[(size_t)b * N_ + row];
    const float wc = expf(s - maxs) * inv_se;
    const float x  = wc * ((row == idx) ? 1.0f : 1e-16f) * invden + 1e-10f;
    const float w  = exp2f(g * log2f(x)) * wnorm;
    if (lane == 0) w_out[(size_t)b * N_ + row] = w;

    const size_t base = ((size_t)b * N_ + row) * M_ + lane * 4;
    float4 c4 = *(const float4*)(content + base);
    float4 k4 = *(const float4*)(keys + base);
    float4 av = *(const float4*)(a   + b * M_ + lane * 4);
    float4 kv = *(const float4*)(a_k + b * K_ + lane * 4);

    float4 nc, nk;
    nc.x = fmaf(w, av.x - c4.x, c4.x);
    nc.y = fmaf(w, av.y - c4.y, c4.y);
    nc.z = fmaf(w, av.z - c4.z, c4.z);
    nc.w = fmaf(w, av.w - c4.w, c4.w);
    nk.x = fmaf(w, kv.x - k4.x, k4.x);
    nk.y = fmaf(w, kv.y - k4.y, k4.y);
    nk.z = fmaf(w, kv.z - k4.z, k4.z);
    nk.w = fmaf(w, kv.w - k4.w, k4.w);

    *(float4*)(nc_out + base) = nc;
    *(float4*)(nk_out + base) = nk;
}

extern "C" void kernel_launch(void* const* d_in, const int* in_sizes, int n_in,
                              void* d_out, int out_size, void* d_ws, size_t ws_size,
                              hipStream_t stream) {
    const float* content = (const float*)d_in[0];  // [B,N,M]
    const float* keys    = (const float*)d_in[1];  // [B,N,K]
    const float* kq      = (const float*)d_in[2];  // [B,K]
    const float* beta    = (const float*)d_in[3];  // [B,1]
    const float* gamma   = (const float*)d_in[4];  // [B,1]
    const float* a_k     = (const float*)d_in[5];  // [B,K]
    const float* a       = (const float*)d_in[6];  // [B,M]

    float* out    = (float*)d_out;
    float* w_out  = out;                                   // B*N
    float* r_out  = w_out + (size_t)B_ * N_;               // B*M
    float* nc_out = r_out + (size_t)B_ * M_;               // B*N*M
    float* nk_out = nc_out + (size_t)B_ * N_ * M_;         // B*N*K

    float* s_buf  = (float*)d_ws;                          // B*N floats (1MB)
    float* params = s_buf + (size_t)B_ * N_;               // B*8 floats

    k1_scores<<<dim3(N_ / 128, B_), 256, 0, stream>>>(keys, kq, beta, s_buf);
    k2_stats <<<dim3(B_), 256, 0, stream>>>(s_buf, content, gamma, params, r_out);
    k3_update<<<dim3(N_ / 8, B_), 256, 0, stream>>>(content, keys, a, a_k, gamma,
                                                    s_buf, params, w_out, nc_out, nk_out);
}